// ERNNCell_75015898792540
// MI455X (gfx1250) — compile-verified
//
#include <hip/hip_runtime.h>
#include <hip/hip_bf16.h>

typedef __attribute__((ext_vector_type(2))) float v2f;
typedef __attribute__((ext_vector_type(4))) float v4f;
typedef __attribute__((ext_vector_type(8))) float v8f;

#define HIDDEN   256
#define NFEAT    128
#define SEQLEN   1024
#define BATCH    256
#define BROWS    16            // batch rows per workgroup (WMMA M)
#define PS       260           // P^T row stride in floats (bank-conflict pad)
#define HSTR     260           // h / A2 row stride
#define XS       132           // x_t row stride

#define PT_SZ    (HIDDEN*PS)   // 66560 floats (P^T)
#define HS_SZ    (BROWS*HSTR)  //  4160 floats (h slice)
#define A2_SZ    (BROWS*HSTR)  //  4160 floats (A2 = 2hP + wx + b staging)
#define XT_SZ    (BROWS*XS)    //  2112 floats (x_t slice)
#define LDS_TOT  (PT_SZ+HS_SZ+A2_SZ+XT_SZ)   // 76992 floats = 307968 B < 320 KB

// D = A(16x4 f32) * B(4x16 f32) + C(16x16 f32), wave32
#define WMMAF32(A,B,C) \
  __builtin_amdgcn_wmma_f32_16x16x4_f32(false,(A),false,(B),(short)0,(C),false,false)

// 16x16 tile += A(16x256, LDS, stride HSTR) * B(256x16 slice of P^T, LDS)
// aptr: &A[m*stride + 2*half]   bptr: &Pt[(n0+m)*PS + 2*half]
__device__ __forceinline__ v8f gemm_k256(const float* aptr, const float* bptr) {
  v8f acc0 = {}, acc1 = {};
#pragma unroll
  for (int kk = 0; kk < 256; kk += 8) {
    v2f a0 = *(const v2f*)(aptr + kk);
    v2f b0 = *(const v2f*)(bptr + kk);
    acc0 = WMMAF32(a0, b0, acc0);
    v2f a1 = *(const v2f*)(aptr + kk + 4);
    v2f b1 = *(const v2f*)(bptr + kk + 4);
    acc1 = WMMAF32(a1, b1, acc1);
  }
  return acc0 + acc1;
}

__global__ __launch_bounds__(512, 1)
void ernn_scan_kernel(const float* __restrict__ x,    // (1024,256,128)
                      const float* __restrict__ h0,   // (256,256)
                      const float* __restrict__ W,    // (128,256)
                      const float* __restrict__ b,    // (256,)
                      const float* __restrict__ V,    // (64,256)
                      const float* __restrict__ V2,   // (256,64)
                      const float* __restrict__ alpha,// (1,1024)
                      const int*   __restrict__ Kp,   // scalar
                      float*       __restrict__ out)  // (256,256)
{
  __shared__ float lds[LDS_TOT];
  float* Pt = lds;                       // P^T[n][k], stride PS
  float* hS = Pt + PT_SZ;                // h slice, stride HSTR
  float* A2 = hS + HS_SZ;                // GEMM2 A operand, stride HSTR
  float* xT = A2 + A2_SZ;                // x_t slice, stride XS

  const int tid  = threadIdx.x;
  const int lane = tid & 31;
  const int wav  = tid >> 5;             // 0..15: owns column tile `wav`
  const int m    = lane & 15;
  const int half = lane >> 4;
  const int r0   = blockIdx.x * BROWS;   // batch row base for this WGP

  // ---- one-time: build P^T = (V2@V + I)^T into LDS ----
  for (int idx = tid; idx < HIDDEN * HIDDEN; idx += 512) {
    const int i  = idx >> 8;             // row of P
    const int jc = idx & 255;            // col of P
    float acc = (i == jc) ? 1.0f : 0.0f;
    const float* v2r = V2 + i * 64;
#pragma unroll 8
    for (int d = 0; d < 64; ++d) acc += v2r[d] * V[d * 256 + jc];
    Pt[jc * PS + i] = acc;               // transposed store
  }
  // ---- one-time: stage h slice ----
  for (int idx = tid; idx < BROWS * 256; idx += 512) {
    const int row = idx >> 8, c = idx & 255;
    hS[row * HSTR + c] = h0[(r0 + row) * 256 + c];
  }
  const float a    = alpha[0];
  const int   Kv   = Kp[0];
  const float bval = b[wav * 16 + m];    // bias for this lane's column
  __syncthreads();

  // per-lane WMMA operand base pointers
  const float* aHp = hS + m * HSTR + 2 * half;            // A from h
  const float* aA2 = A2 + m * HSTR + 2 * half;            // A from A2
  const float* aX  = xT + m * XS   + 2 * half;            // A from x_t
  const float* bP  = Pt + (wav * 16 + m) * PS + 2 * half; // B from P^T
  const float* Wq  = W + (2 * half) * 256 + wav * 16 + m; // B from W (global)
  float* stA2 = A2 + (8 * half) * HSTR + wav * 16 + m;    // C-layout stores
  float* stH  = hS + (8 * half) * HSTR + wav * 16 + m;

  for (int t = 0; t < SEQLEN; ++t) {
    // stage x_t (16x128) : one float4 per thread
    {
      const int row = tid >> 5, c4 = tid & 31;
      const size_t goff = ((size_t)t * BATCH + r0 + row) * NFEAT + c4 * 4;
      *(v4f*)(xT + row * XS + c4 * 4) = *(const v4f*)(x + goff);
      const int tn = (t + 1 < SEQLEN) ? t + 1 : t;      // next-step prefetch
      __builtin_prefetch(x + ((size_t)tn * BATCH + r0 + row) * NFEAT + c4 * 4, 0, 0);
    }
    __syncthreads();                                     // B1

    // GEMM1: hp = h @ P          (16x256x256, f32 WMMA)
    v8f hp8 = gemm_k256(aHp, bP);

    // fused input projection: wv = x_t @ W + b   (16x128x256)
    v8f wv8;
    {
      v8f acc0 = {}, acc1 = {};
#pragma unroll
      for (int kk = 0; kk < NFEAT; kk += 8) {
        v2f a0 = *(const v2f*)(aX + kk);
        v2f b0; b0.x = Wq[kk * 256]; b0.y = Wq[kk * 256 + 256];
        acc0 = WMMAF32(a0, b0, acc0);
        v2f a1 = *(const v2f*)(aX + kk + 4);
        v2f b1; b1.x = Wq[(kk + 4) * 256]; b1.y = Wq[(kk + 4) * 256 + 256];
        acc1 = WMMAF32(a1, b1, acc1);
      }
      wv8 = acc0 + acc1;
#pragma unroll
      for (int r = 0; r < 8; ++r) wv8[r] += bval;
    }
    const v8f av8 = wv8 + hp8;           // alpha_vec tile = wx + b + h@P

    // oldh / h_cur for this wave's own tile (C-layout)
    v8f oldh8;
#pragma unroll
    for (int r = 0; r < 8; ++r) oldh8[r] = stH[r * HSTR];
    v8f cur8 = oldh8;

    for (int k = 0; k < Kv; ++k) {       // uniform trip count (K from input)
      v8f hk8;
      if (k == 0) hk8 = hp8;             // h_cur == h_prev at k==0
      else        hk8 = gemm_k256(aHp, bP);
      const v8f a28 = hk8 + av8;         // A2 = h_cur@P + alpha_vec
#pragma unroll
      for (int r = 0; r < 8; ++r) stA2[r * HSTR] = a28[r];
      __syncthreads();                   // B2: A2 visible to all waves

      // GEMM2: z = A2 @ P
      v8f z8 = gemm_k256(aA2, bP);

      // h_cur = a*(tanh(z) - oldh) + (1-a)*h_cur
#pragma unroll
      for (int r = 0; r < 8; ++r)
        cur8[r] = a * (tanhf(z8[r]) - oldh8[r]) + (1.0f - a) * cur8[r];
#pragma unroll
      for (int r = 0; r < 8; ++r) stH[r * HSTR] = cur8[r];
      __syncthreads();                   // B3: h visible for next GEMM
    }
  }

  // write final h slice
  for (int idx = tid; idx < BROWS * 256; idx += 512) {
    const int row = idx >> 8, c = idx & 255;
    out[(r0 + row) * 256 + c] = hS[row * HSTR + c];
  }
}

extern "C" void kernel_launch(void* const* d_in, const int* in_sizes, int n_in,
                              void* d_out, int out_size, void* d_ws, size_t ws_size,
                              hipStream_t stream) {
  const float* x     = (const float*)d_in[0];
  const float* h     = (const float*)d_in[1];
  const float* W     = (const float*)d_in[2];
  const float* b     = (const float*)d_in[3];
  const float* V     = (const float*)d_in[4];
  const float* V2    = (const float*)d_in[5];
  const float* alpha = (const float*)d_in[6];
  const int*   K     = (const int*)d_in[7];
  float* out = (float*)d_out;

  dim3 grid(BATCH / BROWS);   // 16 independent batch groups -> 16 WGPs
  dim3 block(512);            // 16 wave32s; wave j owns column tile j
  ernn_scan_kernel<<<grid, block, 0, stream>>>(x, h, W, b, V, V2, alpha, K, out);
}